// ConvNetLayer_40312563040424
// MI455X (gfx1250) — compile-verified
//
#include <hip/hip_runtime.h>
#include <hip/hip_bf16.h>
#include <math.h>

#define NN 20000
#define NE 320000

typedef __attribute__((ext_vector_type(2))) float v2f;
typedef __attribute__((ext_vector_type(8))) float v8f;
typedef __attribute__((ext_vector_type(4))) unsigned int u32x4;
typedef __attribute__((ext_vector_type(8))) int i32x8;
typedef __attribute__((ext_vector_type(4))) int i32x4;

#if defined(__has_builtin)
#  if __has_builtin(__builtin_amdgcn_tensor_load_to_lds) && \
      __has_builtin(__builtin_amdgcn_s_wait_tensorcnt)
#    define HAVE_TDM 1
#  endif
#endif

__device__ __forceinline__ v8f wmma4(v2f a, v2f b, v8f c) {
  // D = A(16x4) * B(4x16) + C(16x16), fp32, wave32
  return __builtin_amdgcn_wmma_f32_16x16x4_f32(false, a, false, b, (short)0, c, false, false);
}

#define INV_MUL   0.125f                 // 1/sqrt(64)
#define INV_UV    0.04419417382415922f   // 1/sqrt(64*8)
#define INV_B8    0.3535533905932738f    // 1/sqrt(8)
#define INV_H64   0.125f                 // 1/sqrt(64)
#define INV_NN    0.25f                  // 1/sqrt(16)
#define INV_MID   0.08838834764831845f   // 1/sqrt(128)
#define INV_SQRT3 0.5773502691896258f
#define LOG2_C    0.6931471805599453f

// ---------------------------------------------------------------------------
// Kernel 1: per 16-node tile. Raw node_feat tile (16x256 f32, contiguous 16KB)
// is DMA'd into LDS by the Tensor Data Mover; the v-component de-interleave
// (64 + u*3 + c) is folded into the WMMA A-fragment addressing.
//   s    = s0@W1s*inv_mul               -> s_ws  (N,64)
//   v_c  = v0_c@W1v*inv_mul             -> v_ws  (N,64,3) flat [u*3+c]
//   sc_s = (s0  x attr)@Wsc_s*inv_uv    -> sc_ws (N,256) cols [0,64)
//   sc_v = (v0_c x attr)@Wsc_v*inv_uv   -> sc_ws cols 64+w*3+c
// 32 output tiles of 16x16 per block; 8 waves x 4 reps.
// ---------------------------------------------------------------------------
__global__ __launch_bounds__(256) void node_prep_kernel(
    const float* __restrict__ node_feat, const float* __restrict__ node_attr,
    const float* __restrict__ W1s, const float* __restrict__ W1v,
    const float* __restrict__ Wsc_s, const float* __restrict__ Wsc_v,
    float* __restrict__ s_ws, float* __restrict__ v_ws, float* __restrict__ sc_ws)
{
  __shared__ float fL[16*256];    // raw node_feat tile, row-major
  __shared__ float aL[16*8];      // attr tile
  const int t  = threadIdx.x;
  const int n0 = blockIdx.x * 16;

#if defined(HAVE_TDM)
  if (t < 32) {   // one wave issues the tensor DMA (EXEC-independent, per-wave)
    unsigned long long ga =
        (unsigned long long)(const void*)(node_feat + (size_t)n0*256);
    unsigned int lds = (unsigned int)(unsigned long long)(void*)fL;
    u32x4 g0;
    g0[0] = 1u;                                    // count=1 (valid user D#)
    g0[1] = lds;                                   // lds_addr [63:32]
    g0[2] = (unsigned int)(ga & 0xFFFFFFFFu);      // global_addr[31:0]
    g0[3] = (unsigned int)((ga >> 32) & 0x01FFFFFFu) | (2u << 30); // addr[56:32]|type=2
    unsigned long long td0 = (unsigned long long)NN * 256ull;      // tensor_dim0
    i32x8 g1;
    g1[0] = (int)(2u << 16);                       // data_size=2 (4 bytes)
    g1[1] = (int)((td0 & 0xFFFFu) << 16);          // tensor_dim0[15:0] @ bits[63:48]
    g1[2] = (int)(((td0 >> 16) & 0xFFFFu) | (1u << 16)); // dim0[31:16] | tensor_dim1=1
    g1[3] = (int)(4096u << 16);                    // tile_dim0 = 4096 elems (1-D tile)
    g1[4] = 0;                                     // tile_dim1 = tile_dim2 = 0
    g1[5] = 256;                                   // tensor_dim0_stride (elements)
    g1[6] = 0;
    g1[7] = 0;
    i32x4 gz = {0,0,0,0};
#  if defined(__clang_major__) && __clang_major__ >= 23
    i32x8 gz8 = {0,0,0,0,0,0,0,0};
    __builtin_amdgcn_tensor_load_to_lds(g0, g1, gz, gz, gz8, 0);
#  else
    __builtin_amdgcn_tensor_load_to_lds(g0, g1, gz, gz, 0);
#  endif
    __builtin_amdgcn_s_wait_tensorcnt(0);
  }
#else
  for (int i = t; i < 4096; i += 256)              // contiguous fallback stage
    fL[i] = node_feat[(size_t)n0*256 + i];
#endif
  for (int i = t; i < 128; i += 256)
    aL[i] = node_attr[(size_t)n0*8 + i];           // 16 rows contiguous
  __syncthreads();

  const int lane = t & 31, wv = t >> 5;
  const int m = lane & 15, hl = lane >> 4;

  for (int rep = 0; rep < 4; ++rep) {
    const int  tile   = wv + rep*8;
    const bool is_sc  = tile >= 16;
    const int  tt     = is_sc ? tile - 16 : tile;     // 0..15
    const bool is_s   = tt < 4;
    const int  c      = is_s ? 0 : ((tt - 4) >> 2);
    const int  colbase= is_s ? tt*16 : ((tt - 4) & 3)*16;
    // A-fragment addressing into the raw tile: s0 -> cols [0,64); v_c -> 64+u*3+c
    const float* Ab   = fL + m*256 + (is_s ? 0 : 64 + c);
    const int    astr = is_s ? 1 : 3;
    const float* B    = is_sc ? (is_s ? Wsc_s : Wsc_v) : (is_s ? W1s : W1v);

    v8f acc = {0.f,0.f,0.f,0.f,0.f,0.f,0.f,0.f};
    if (!is_sc) {                                  // linear GEMM, K=64
      for (int k0 = 0; k0 < 64; k0 += 4) {
        int ka = k0 + 2*hl;
        v2f a, b;
        a.x = Ab[ka*astr];                a.y = Ab[(ka+1)*astr];
        b.x = B[ka*64 + colbase + m];     b.y = B[(ka+1)*64 + colbase + m];
        acc = wmma4(a, b, acc);
      }
    } else {                                       // self-connection GEMM, K=512
      for (int k0 = 0; k0 < 512; k0 += 4) {
        int ka = k0 + 2*hl, kb = ka + 1;
        v2f a, b;
        a.x = Ab[(ka>>3)*astr] * aL[m*8 + (ka&7)]; // on-the-fly outer product
        a.y = Ab[(kb>>3)*astr] * aL[m*8 + (kb&7)];
        b.x = B[ka*64 + colbase + m];
        b.y = B[kb*64 + colbase + m];
        acc = wmma4(a, b, acc);
      }
    }

    // Branchless store: wave-uniform pointer/pitch select, single store path.
    const float scale = is_sc ? INV_UV : INV_MUL;
    const int   gn0   = n0 + hl*8;
    const int   col   = colbase + m;
    float* dp; int pitch;
    if (is_sc) { dp = sc_ws + (size_t)gn0*256 + (is_s ? col : 64 + col*3 + c); pitch = 256; }
    else if (is_s) { dp = s_ws + (size_t)gn0*64 + col;               pitch = 64;  }
    else           { dp = v_ws + (size_t)gn0*192 + col*3 + c;        pitch = 192; }
    #pragma unroll
    for (int r = 0; r < 8; ++r)
      dp[(size_t)r * pitch] = acc[r] * scale;
  }
}

// ---------------------------------------------------------------------------
// Kernel 2: per 16-edge tile.  MLP (WMMA for the K=64 stage) -> messages ->
// atomic scatter into agg_s (N,128) / agg_v (N,128,3). agg fits in L2 (41MB),
// so the f32 atomic storm resolves on-chip.
// ---------------------------------------------------------------------------
__global__ __launch_bounds__(256) void edge_kernel(
    const float* __restrict__ edge_sh, const float* __restrict__ edge_basis,
    const float* __restrict__ Wfc1, const float* __restrict__ Wfc2,
    const int* __restrict__ edge_idx,
    const float* __restrict__ s_ws, const float* __restrict__ v_ws,
    float* __restrict__ agg_s, float* __restrict__ agg_v)
{
  __shared__ float bL[16*8];
  __shared__ float hL[16*64];
  __shared__ float wL[16*256];
  const int t  = threadIdx.x;
  const int e0 = blockIdx.x * 16;

  if (t < 128) bL[t] = edge_basis[(size_t)e0*8 + t];   // 16 rows contiguous
  __syncthreads();

  { // h = ssp(basis @ Wfc1 * inv8): 1024 values, 4 per thread
    int e = t & 15, j0 = (t >> 4) * 4;
    for (int j = j0; j < j0+4; ++j) {
      float acc = 0.f;
      #pragma unroll
      for (int b = 0; b < 8; ++b) acc += bL[e*8+b] * Wfc1[b*64 + j];
      float x  = acc * INV_B8;
      float sp = fmaxf(x, 0.f) + log1pf(__expf(-fabsf(x)));  // stable softplus
      hL[e*64 + j] = sp - LOG2_C;
    }
  }
  __syncthreads();

  const int lane = t & 31, wv = t >> 5;
  const int m = lane & 15, hl = lane >> 4;
  for (int sub = 0; sub < 2; ++sub) {      // w = h @ Wfc2 * inv64 (K=64)
    int colbase = (wv*2 + sub) * 16;
    v8f acc = {0.f,0.f,0.f,0.f,0.f,0.f,0.f,0.f};
    for (int k0 = 0; k0 < 64; k0 += 4) {
      int ka = k0 + 2*hl;
      v2f a, b;
      a.x = hL[m*64 + ka];                  a.y = hL[m*64 + ka + 1];
      b.x = Wfc2[ka*256 + colbase + m];     b.y = Wfc2[(ka+1)*256 + colbase + m];
      acc = wmma4(a, b, acc);
    }
    #pragma unroll
    for (int r = 0; r < 8; ++r)
      wL[(hl*8 + r)*256 + colbase + m] = acc[r] * INV_H64;
  }
  __syncthreads();

  { // gather + message + scatter: thread t -> edge e=t>>4, channels u0..u0+3
    int e = t >> 4, u0 = (t & 15) * 4;
    int ge  = e0 + e;
    int dst = edge_idx[(size_t)ge*2 + 0];
    int src = edge_idx[(size_t)ge*2 + 1];
    float sh0  = edge_sh[(size_t)ge*4 + 0];
    float sh1x = edge_sh[(size_t)ge*4 + 1];
    float sh1y = edge_sh[(size_t)ge*4 + 2];
    float sh1z = edge_sh[(size_t)ge*4 + 3];
    float* aggs = agg_s + (size_t)dst*128;
    float* aggv = agg_v + (size_t)dst*384;
    #pragma unroll
    for (int uu = 0; uu < 4; ++uu) {
      int u = u0 + uu;
      float xs  = s_ws[(size_t)src*64 + u];
      float xv0 = v_ws[(size_t)src*192 + u*3 + 0];
      float xv1 = v_ws[(size_t)src*192 + u*3 + 1];
      float xv2 = v_ws[(size_t)src*192 + u*3 + 2];
      float wA = wL[e*256 + u];
      float wB = wL[e*256 + 64  + u];
      float wC = wL[e*256 + 128 + u];
      float wD = wL[e*256 + 192 + u];
      unsafeAtomicAdd(aggs + u, wA * xs * sh0);
      float dv = xv0*sh1x + xv1*sh1y + xv2*sh1z;
      unsafeAtomicAdd(aggs + 64 + u, wD * dv * INV_SQRT3);
      float bxs = wB * xs;
      unsafeAtomicAdd(aggv + u*3 + 0, bxs * sh1x);
      unsafeAtomicAdd(aggv + u*3 + 1, bxs * sh1y);
      unsafeAtomicAdd(aggv + u*3 + 2, bxs * sh1z);
      float csh = wC * sh0;
      unsafeAtomicAdd(aggv + (64+u)*3 + 0, csh * xv0);
      unsafeAtomicAdd(aggv + (64+u)*3 + 1, csh * xv1);
      unsafeAtomicAdd(aggv + (64+u)*3 + 2, csh * xv2);
    }
  }
}

// ---------------------------------------------------------------------------
// Kernel 3: out = agg @ W2 * (inv_nn*inv_mid) + sc, per 16-node tile (K=128).
// Single branchless WMMA loop: pointer/stride select handles s vs v tiles.
// ---------------------------------------------------------------------------
__global__ __launch_bounds__(256) void out_kernel(
    const float* __restrict__ agg_s, const float* __restrict__ agg_v,
    const float* __restrict__ W2s, const float* __restrict__ W2v,
    const float* __restrict__ sc_ws, float* __restrict__ out)
{
  __shared__ float sA[16*128];
  __shared__ float vA[16*384];
  const int t  = threadIdx.x;
  const int n0 = blockIdx.x * 16;
  for (int i = t; i < 2048; i += 256) sA[i] = agg_s[(size_t)n0*128 + i];
  for (int i = t; i < 6144; i += 256) vA[i] = agg_v[(size_t)n0*384 + i];
  __syncthreads();

  const int lane = t & 31, wv = t >> 5;
  const int m = lane & 15, hl = lane >> 4;
  const float scale = INV_NN * INV_MID;

  for (int rep = 0; rep < 2; ++rep) {
    const int  tile   = wv + rep*8;
    const bool is_s   = tile < 4;
    const int  tt     = is_s ? tile : tile - 4;
    const int  c      = is_s ? 0 : (tt >> 2);
    const int  colbase= is_s ? tt*16 : (tt & 3)*16;
    const float* Ab   = is_s ? (sA + m*128) : (vA + m*384 + c);
    const int    astr = is_s ? 1 : 3;
    const float* B    = is_s ? W2s : W2v;

    v8f acc = {0.f,0.f,0.f,0.f,0.f,0.f,0.f,0.f};
    for (int k0 = 0; k0 < 128; k0 += 4) {
      int ka = k0 + 2*hl;
      v2f a, b;
      a.x = Ab[ka*astr];                a.y = Ab[(ka+1)*astr];
      b.x = B[ka*64 + colbase + m];     b.y = B[(ka+1)*64 + colbase + m];
      acc = wmma4(a, b, acc);
    }

    const int baseoff = is_s ? (colbase + m) : (64 + (colbase + m)*3 + c);
    #pragma unroll
    for (int r = 0; r < 8; ++r) {
      size_t idx = (size_t)(n0 + hl*8 + r)*256 + baseoff;
      out[idx] = acc[r]*scale + sc_ws[idx];
    }
  }
}

// ---------------------------------------------------------------------------
extern "C" void kernel_launch(void* const* d_in, const int* in_sizes, int n_in,
                              void* d_out, int out_size, void* d_ws, size_t ws_size,
                              hipStream_t stream) {
  const float* node_feat  = (const float*)d_in[0];
  const float* node_attr  = (const float*)d_in[1];
  const float* edge_sh    = (const float*)d_in[2];
  const float* edge_basis = (const float*)d_in[3];
  const float* W1s   = (const float*)d_in[4];
  const float* W1v   = (const float*)d_in[5];
  const float* Wfc1  = (const float*)d_in[6];
  const float* Wfc2  = (const float*)d_in[7];
  const float* W2s   = (const float*)d_in[8];
  const float* W2v   = (const float*)d_in[9];
  const float* Wsc_s = (const float*)d_in[10];
  const float* Wsc_v = (const float*)d_in[11];
  const int*   edge_idx = (const int*)d_in[12];

  // workspace layout (floats): s | v | sc | agg_s | agg_v  = N*1024 floats
  float* ws    = (float*)d_ws;
  float* s_ws  = ws;
  float* v_ws  = ws + (size_t)NN*64;
  float* sc_ws = ws + (size_t)NN*256;
  float* agg_s = ws + (size_t)NN*512;
  float* agg_v = ws + (size_t)NN*640;

  hipMemsetAsync(agg_s, 0, (size_t)NN*512*sizeof(float), stream);  // agg_s+agg_v
  node_prep_kernel<<<NN/16, 256, 0, stream>>>(node_feat, node_attr, W1s, W1v,
                                              Wsc_s, Wsc_v, s_ws, v_ws, sc_ws);
  edge_kernel<<<NE/16, 256, 0, stream>>>(edge_sh, edge_basis, Wfc1, Wfc2,
                                         edge_idx, s_ws, v_ws, agg_s, agg_v);
  out_kernel<<<NN/16, 256, 0, stream>>>(agg_s, agg_v, W2s, W2v, sc_ws,
                                        (float*)d_out);
}